// Def_A2C_Sample_Generator_13761075216640
// MI455X (gfx1250) — compile-verified
//
#include <hip/hip_runtime.h>
#include <hip/hip_bf16.h>

#define T_N   500
#define R_N   50
#define NS_N  1000
#define H1_N  32
#define H2_N  16

typedef __attribute__((ext_vector_type(2))) float v2f;
typedef __attribute__((ext_vector_type(8))) float v8f;

// ---------------------------------------------------------------------------
// K1: xw1[t][c] = sum_f x[t][f] * gc1_w[f][c], x = cat(state, payoff_row, noise)
// grid = T, block = 32 (one wave). payoff reads are lane-uniform broadcasts,
// w1 reads are coalesced across lanes.
// ---------------------------------------------------------------------------
__global__ __launch_bounds__(32)
void k_xw1(const float* __restrict__ state, const float* __restrict__ payoff,
           const float* __restrict__ noise, const float* __restrict__ w1,
           float* __restrict__ xw1) {
  const int t = blockIdx.x;
  const int c = threadIdx.x;  // 0..31
  float acc = state[t * 2 + 0] * w1[0 * H1_N + c]
            + state[t * 2 + 1] * w1[1 * H1_N + c];
  for (int j = 0; j < T_N; ++j)
    acc += payoff[t * T_N + j] * w1[(2 + j) * H1_N + c];
  acc += noise[t * 2 + 0] * w1[502 * H1_N + c];
  acc += noise[t * 2 + 1] * w1[503 * H1_N + c];
  xw1[t * H1_N + c] = acc;
}

// ---------------------------------------------------------------------------
// WMMA fp32 GEMM: C = A(MxK) @ B(KxN), row-major, one wave per 16x16 C tile.
// Uses V_WMMA_F32_16X16X4_F32 (fp32 matrix path on gfx1250).
//   A 16x4 layout: lanes 0-15 -> M=0..15, VGPR0/1 = K pair (koff = 2*(lane>=16))
//   B 4x16 layout: mirrored; C/D: n = lane&15, M = vgpr + 8*(lane>=16)
// Out-of-tile rows/cols are handled branch-free: indices are clamped so loads
// stay in-bounds and the loaded values are multiplied by a 0/1 mask, keeping
// the inner loop free of EXEC save/restore sequences.
// EPI 0 (GCN): out = lrelu((acc + bias[n]) * gamma[m] + beta[m])
// EPI 1 (head): out = acc + addn[n]
// ---------------------------------------------------------------------------
template <int EPI>
__global__ __launch_bounds__(256)
void k_wmma_gemm(const float* __restrict__ A, const float* __restrict__ B,
                 float* __restrict__ C, int M, int N, int K,
                 const float* __restrict__ bias, const float* __restrict__ gamma,
                 const float* __restrict__ beta, const float* __restrict__ addn) {
  const int lane  = threadIdx.x & 31;
  const int wave  = blockIdx.x * (blockDim.x >> 5) + (threadIdx.x >> 5);
  const int tilesN = (N + 15) >> 4;
  const int tilesM = (M + 15) >> 4;
  if (wave >= tilesM * tilesN) return;           // wave-uniform: EXEC stays full
  const int m0 = (wave / tilesN) << 4;
  const int n0 = (wave % tilesN) << 4;
  const int mr   = lane & 15;                    // A row in tile / C column idx
  const int koff = (lane >> 4) << 1;             // 0 or 2
  const int col  = n0 + mr;                      // B / C column
  const int arow = m0 + mr;                      // A row

  const float amask = (arow < M) ? 1.f : 0.f;
  const float bmask = (col  < N) ? 1.f : 0.f;
  const float* __restrict__ Arow = A + (size_t)(arow < M ? arow : 0) * K + koff;
  const float* __restrict__ Bcol = B + (col < N ? col : 0) + (size_t)koff * N;

  v8f acc = {};
#pragma unroll 4
  for (int kk = 0; kk < K; kk += 4) {
    // A pair is contiguous & 8B-aligned (kk+koff is even) -> global_load_b64
    const v2f araw = *(const v2f*)(Arow + kk);
    v2f a, b;
    a.x = araw.x * amask;
    a.y = araw.y * amask;
    b.x = Bcol[(size_t)kk * N]       * bmask;
    b.y = Bcol[(size_t)(kk + 1) * N] * bmask;
    acc = __builtin_amdgcn_wmma_f32_16x16x4_f32(
        /*neg_a=*/false, a, /*neg_b=*/false, b,
        /*c_mod=*/(short)0, acc, /*reuse_a=*/false, /*reuse_b=*/false);
  }

  const int mbase = m0 + ((lane >> 4) << 3);     // +8 for upper lane half
  if (col < N) {
#pragma unroll
    for (int i = 0; i < 8; ++i) {
      const int m = mbase + i;
      if (m < M) {
        float v = acc[i];
        if (EPI == 0) {
          v = (v + bias[col]) * gamma[m] + beta[m];
          v = v >= 0.f ? v : 0.2f * v;           // LeakyReLU(0.2)
        } else {
          v = v + addn[col];
        }
        C[m * N + col] = v;
      }
    }
  }
}

// ---------------------------------------------------------------------------
// K3: hw2[t][c] = sum_j h1[t][j] * gc2_w[j][c]  (500x32x16, trivial)
// ---------------------------------------------------------------------------
__global__ __launch_bounds__(256)
void k_hw2(const float* __restrict__ h1, const float* __restrict__ w2,
           float* __restrict__ hw2) {
  const int id = blockIdx.x * blockDim.x + threadIdx.x;
  if (id >= T_N * H2_N) return;
  const int t = id >> 4, c = id & 15;
  float acc = 0.f;
#pragma unroll
  for (int j = 0; j < H1_N; ++j) acc += h1[t * H1_N + j] * w2[j * H2_N + c];
  hw2[id] = acc;
}

// ---------------------------------------------------------------------------
// K4: lw[t] = sum_k h2_flat[k] * actgen_w[k][t]   (xf == h2 row-major flat)
// consecutive threads read consecutive columns -> coalesced per k.
// ---------------------------------------------------------------------------
__global__ __launch_bounds__(256)
void k_lw(const float* __restrict__ h2, const float* __restrict__ W,
          float* __restrict__ lw) {
  const int t = blockIdx.x * blockDim.x + threadIdx.x;
  if (t >= T_N) return;
  float acc = 0.f;
  for (int k = 0; k < H2_N * T_N; ++k) acc += h2[k] * W[k * T_N + t];
  lw[t] = acc;
}

// ---------------------------------------------------------------------------
// K5 (the memory-bound 200MB streamer): one wave32 per (sample,resource) row.
// Row of 500 kept in registers (16 f32/lane): gumbel + stable softmax in a
// single read / single write pass. Wave reductions via __shfl_xor (wave32).
// gumbel_u / out are touched exactly once -> non-temporal (TH=NT) so the hot
// logits rows (read 1000x each) keep their residency in L2.
// ---------------------------------------------------------------------------
__global__ __launch_bounds__(256)
void k_gumbel_softmax(const float* __restrict__ u,
                      const float* __restrict__ logits,
                      float* __restrict__ out) {
  const int lane = threadIdx.x & 31;
  const int row  = blockIdx.x * (blockDim.x >> 5) + (threadIdx.x >> 5);
  const int r    = row % R_N;
  const float* __restrict__ lrow = logits + r * T_N;
  const float* __restrict__ urow = u + (size_t)row * T_N;
  float* __restrict__ orow = out + (size_t)row * T_N;

  float z[16];
  float mx = -1e30f;
#pragma unroll
  for (int j = 0; j < 16; ++j) {
    const int idx = lane + (j << 5);
    if (idx < T_N) {
      const float uu = __builtin_nontemporal_load(urow + idx);
      const float g = -__logf(-__logf(uu));
      const float v = lrow[idx] + g;
      z[j] = v;
      mx = fmaxf(mx, v);
    } else {
      z[j] = -1e30f;
    }
  }
#pragma unroll
  for (int off = 16; off > 0; off >>= 1)
    mx = fmaxf(mx, __shfl_xor(mx, off, 32));

  float s = 0.f;
#pragma unroll
  for (int j = 0; j < 16; ++j) {
    const float e = __expf(z[j] - mx);   // padded lanes -> exp(-huge) = 0
    z[j] = e;
    s += e;
  }
#pragma unroll
  for (int off = 16; off > 0; off >>= 1)
    s += __shfl_xor(s, off, 32);

  const float inv = 1.f / s;
#pragma unroll
  for (int j = 0; j < 16; ++j) {
    const int idx = lane + (j << 5);
    if (idx < T_N) __builtin_nontemporal_store(z[j] * inv, orow + idx);
  }
}

// ---------------------------------------------------------------------------
extern "C" void kernel_launch(void* const* d_in, const int* in_sizes, int n_in,
                              void* d_out, int out_size, void* d_ws, size_t ws_size,
                              hipStream_t stream) {
  const float* state  = (const float*)d_in[0];
  const float* dcl    = (const float*)d_in[1];   // def_cur_loc (50,500)
  const float* noise  = (const float*)d_in[2];
  const float* gum    = (const float*)d_in[3];   // (1000,50,500)
  const float* payoff = (const float*)d_in[4];
  const float* adj    = (const float*)d_in[5];   // norm_adj (500,500)
  const float* w1     = (const float*)d_in[6];
  const float* b1     = (const float*)d_in[7];
  const float* w2     = (const float*)d_in[8];
  const float* b2     = (const float*)d_in[9];
  const float* gamma  = (const float*)d_in[10];
  const float* beta   = (const float*)d_in[11];
  const float* aw     = (const float*)d_in[12];  // actgen_w (8000,500)
  const float* av     = (const float*)d_in[13];  // actgen_v (500,500)
  float* out = (float*)d_out;

  float* ws     = (float*)d_ws;
  float* xw1    = ws;            // 500*32 = 16000
  float* h1     = ws + 16000;    // 500*32
  float* hw2    = ws + 32000;    // 500*16 = 8000
  float* h2     = ws + 40000;    // 500*16
  float* lw     = ws + 48000;    // 500 (+pad)
  float* logits = ws + 48512;    // 50*500 = 25000

  // front-end
  k_xw1<<<T_N, 32, 0, stream>>>(state, payoff, noise, w1, xw1);
  // GCN1: adj(500x500) @ xw1(500x32) -> 32x2 = 64 tiles = 8 blocks * 8 waves
  k_wmma_gemm<0><<<8, 256, 0, stream>>>(adj, xw1, h1, T_N, H1_N, T_N,
                                        b1, gamma, beta, nullptr);
  k_hw2<<<(T_N * H2_N + 255) / 256, 256, 0, stream>>>(h1, w2, hw2);
  // GCN2: adj @ hw2(500x16) -> 32x1 = 32 tiles = 4 blocks * 8 waves
  k_wmma_gemm<0><<<4, 256, 0, stream>>>(adj, hw2, h2, T_N, H2_N, T_N,
                                        b2, gamma, beta, nullptr);
  k_lw<<<2, 256, 0, stream>>>(h2, aw, lw);
  // logits = dcl(50x500) @ av(500x500) + lw[n] -> 4x32 = 128 tiles = 16 blocks
  k_wmma_gemm<1><<<16, 256, 0, stream>>>(dcl, av, logits, R_N, T_N, T_N,
                                         nullptr, nullptr, nullptr, lw);
  // 50,000 rows, one wave each, 8 waves/block
  k_gumbel_softmax<<<(NS_N * R_N) / 8, 256, 0, stream>>>(gum, logits, out);
}